// BaselineDNN_47132971106337
// MI455X (gfx1250) — compile-verified
//
#include <hip/hip_runtime.h>
#include <hip/hip_bf16.h>

typedef __attribute__((ext_vector_type(2))) float v2f;
typedef __attribute__((ext_vector_type(8))) float v8f;

#define B_  2048
#define L_  200
#define D_  300
#define H_  32
#define C_  3

// ---------------------------------------------------------------------------
// Kernel 1: embedding gather + length-normalized sum.
// One block per batch row; 320 threads (10 wave32s), lane t owns column t
// (t < 300). Indices staged in LDS so the 200-deep loop has no dependent
// global index fetch; global_prefetch_b8 pulls rows ahead into cache.
// Memory-bound: ~491 MB total, table (60 MB) resident in 192 MB L2.
// ---------------------------------------------------------------------------
__global__ __launch_bounds__(320) void embsum_kernel(
    const int* __restrict__ x, const int* __restrict__ lengths,
    const float* __restrict__ emb, float* __restrict__ rep)
{
    __shared__ int idx[L_];
    const int b   = blockIdx.x;
    const int tid = threadIdx.x;

    const int* xr = x + (long long)b * L_;
    for (int l = tid; l < L_; l += 320) idx[l] = xr[l];
    __syncthreads();

    if (tid >= D_) return;

    float acc = 0.f;
    #pragma unroll 2
    for (int l = 0; l < L_; ++l) {
        const long long row = (long long)idx[l] * D_;
        if (l + 8 < L_) {
            // gfx1250 global_prefetch_b8 for the row 8 iterations ahead
            __builtin_prefetch(&emb[(long long)idx[l + 8] * D_ + tid], 0, 1);
        }
        acc += emb[row + tid];
    }

    rep[(long long)b * D_ + tid] = acc / (float)lengths[b];
}

// ---------------------------------------------------------------------------
// Kernel 2: logits = relu(rep @ W1 + b1) @ W2 + b2 using V_WMMA_F32_16X16X4_F32.
// One wave (32 threads) per block, 16 batch rows per block, two 16x16 N-tiles
// (H=32). K=300 stepped by 4 -> 75 WMMA pairs, f32 accumulation (exact dtype).
// A 16x4 layout: lane = {M = lane&15, K-pair = (lane>>4)*2}; B 4x16 mirrored.
// C/D layout: VGPR r, lane -> M = r + 8*(lane>>4), N = lane&15.
// ---------------------------------------------------------------------------
__global__ __launch_bounds__(32) void mlp_wmma_kernel(
    const float* __restrict__ rep,
    const float* __restrict__ W1, const float* __restrict__ b1,
    const float* __restrict__ W2, const float* __restrict__ b2,
    float* __restrict__ out)
{
    __shared__ float hbuf[16 * H_];

    const int lane = threadIdx.x;        // 0..31, full wave (EXEC all ones)
    const int m0   = blockIdx.x * 16;
    const int m    = lane & 15;          // A row owned by this lane
    const int kk   = (lane >> 4) * 2;    // A/B K-pair select (0 or 2)
    const int n    = lane & 15;          // B/C column within tile

    v8f c0 = {};   // N = 0..15
    v8f c1 = {};   // N = 16..31

    const float* arow = rep + (long long)(m0 + m) * D_;

    for (int k = 0; k < D_; k += 4) {
        v2f a;
        const float* ap = arow + k + kk;
        a.x = ap[0];
        a.y = ap[1];

        const float* wp = W1 + (long long)(k + kk) * H_;
        v2f bv0, bv1;
        bv0.x = wp[n];            // K = k+kk,   N = n
        bv0.y = wp[H_ + n];       // K = k+kk+1, N = n
        bv1.x = wp[16 + n];       // K = k+kk,   N = 16+n
        bv1.y = wp[H_ + 16 + n];  // K = k+kk+1, N = 16+n

        c0 = __builtin_amdgcn_wmma_f32_16x16x4_f32(
                 false, a, false, bv0, (short)0, c0, false, false);
        c1 = __builtin_amdgcn_wmma_f32_16x16x4_f32(
                 false, a, false, bv1, (short)0, c1, false, false);
    }

    // h = relu(c + b1) -> LDS  (C/D layout: M = r + 8*(lane>>4), N = lane&15)
    const int mbase = (lane >> 4) * 8;
    const float bb0 = b1[n];
    const float bb1 = b1[16 + n];
    #pragma unroll
    for (int r = 0; r < 8; ++r) {
        const int mm = mbase + r;
        float h0 = c0[r] + bb0; h0 = h0 > 0.f ? h0 : 0.f;
        float h1 = c1[r] + bb1; h1 = h1 > 0.f ? h1 : 0.f;
        hbuf[mm * H_ + n]      = h0;
        hbuf[mm * H_ + 16 + n] = h1;
    }
    __syncthreads();

    // logits = h @ W2 + b2  (16 rows x 3 cols, trivial)
    if (lane < 16) {
        float a0 = b2[0], a1 = b2[1], a2 = b2[2];
        #pragma unroll
        for (int nn = 0; nn < H_; ++nn) {
            const float hv = hbuf[lane * H_ + nn];
            a0 += hv * W2[nn * C_ + 0];
            a1 += hv * W2[nn * C_ + 1];
            a2 += hv * W2[nn * C_ + 2];
        }
        float* o = out + (long long)(m0 + lane) * C_;
        o[0] = a0; o[1] = a1; o[2] = a2;
    }
}

extern "C" void kernel_launch(void* const* d_in, const int* in_sizes, int n_in,
                              void* d_out, int out_size, void* d_ws, size_t ws_size,
                              hipStream_t stream)
{
    const int*   x       = (const int*)  d_in[0];   // [B, L] int32
    const int*   lengths = (const int*)  d_in[1];   // [B]    int32
    const float* emb     = (const float*)d_in[2];   // [V, D]
    const float* W1      = (const float*)d_in[3];   // [D, H]
    const float* b1      = (const float*)d_in[4];   // [H]
    const float* W2      = (const float*)d_in[5];   // [H, C]
    const float* b2      = (const float*)d_in[6];   // [C]

    float* rep = (float*)d_ws;                      // [B, D] scratch (2.4 MB)
    float* out = (float*)d_out;                     // [B, C]

    embsum_kernel<<<B_, 320, 0, stream>>>(x, lengths, emb, rep);
    mlp_wmma_kernel<<<B_ / 16, 32, 0, stream>>>(rep, W1, b1, W2, b2, out);
}